// LSHSelfAttention_44762149159258
// MI455X (gfx1250) — compile-verified
//
#include <hip/hip_runtime.h>
#include <math.h>

// ---------------------------------------------------------------------------
// LSH self-attention (Reformer) for gfx1250, all-fp32 via V_WMMA_F32_16X16X4_F32.
// Pipeline: proj GEMM -> LSH hash GEMM+argmax -> stable counting sort ->
//           chunked attention (WMMA dots + WMMA PV, fused unsort scatter) ->
//           round-softmax combine.
// ---------------------------------------------------------------------------

typedef __attribute__((ext_vector_type(2))) float f32x2;
typedef __attribute__((ext_vector_type(8))) float f32x8;

#define kB        2
#define kH        8
#define kS        4096
#define kD        128
#define kHid      1024
#define kNH       4
#define kNSort    (kNH * kS)       // 16384 sorted slots per (b,h)
#define kNChunks  (kNSort / 64)    // 256 chunks per (b,h)
#define GP        132              // padded LDS row pitch (floats)

__device__ __forceinline__ f32x8 wmma_f32(f32x2 a, f32x2 b, f32x8 c) {
  // D = A(16x4) * B(4x16) + C(16x16), fp32
  return __builtin_amdgcn_wmma_f32_16x16x4_f32(false, a, false, b, (short)0, c,
                                               false, false);
}

// ---------------------------------------------------------------------------
// K1: qk = X @ Wqk^T, v = X @ Wv^T ; output layout [B,H,S,D]
// One wave computes a 64x64 tile (4x4 WMMA C tiles): 8 b64-loads : 16 WMMA
// per k-step. 4096 wave-tiles total.
// ---------------------------------------------------------------------------
__global__ __launch_bounds__(256) void proj_kernel(
    const float* __restrict__ X, const float* __restrict__ Wqk,
    const float* __restrict__ Wv, float* __restrict__ qk_out,
    float* __restrict__ v_out) {
  const int lane = threadIdx.x & 31;
  const int wave = threadIdx.x >> 5;
  const int lm = lane & 15, hf = lane >> 4;
  const int NT = kHid / 64;        // 16
  const int MT = (kB * kS) / 64;   // 128
  int tile = blockIdx.x * 8 + wave;
  int which = tile / (MT * NT);
  int rem = tile % (MT * NT);
  int mt = rem / NT, nt = rem % NT;
  const float* W = which ? Wv : Wqk;
  float* Out = which ? v_out : qk_out;
  const int m0 = mt * 64, n0 = nt * 64;

  f32x8 acc[16] = {};  // [mi*4+ni]
  // Per-lane fragment pointers: lane needs the contiguous pair at k + hf*2.
  const f32x2* ap[4];
  const f32x2* bp[4];
  #pragma unroll
  for (int i = 0; i < 4; ++i) {
    ap[i] = (const f32x2*)(X + (size_t)(m0 + i * 16 + lm) * kHid + hf * 2);
    bp[i] = (const f32x2*)(W + (size_t)(n0 + i * 16 + lm) * kHid + hf * 2);
  }

  for (int k = 0; k < kHid; k += 4) {
    f32x2 a[4], b[4];
    #pragma unroll
    for (int i = 0; i < 4; ++i) {
      a[i] = ap[i][k >> 1];   // floats [k + hf*2, k + hf*2 + 1]
      b[i] = bp[i][k >> 1];
    }
    #pragma unroll
    for (int i = 0; i < 4; ++i)
      #pragma unroll
      for (int j = 0; j < 4; ++j)
        acc[i * 4 + j] = wmma_f32(a[i], b[j], acc[i * 4 + j]);
  }
  // scatter into [B,H,S,D]
  #pragma unroll
  for (int mi = 0; mi < 4; ++mi) {
    #pragma unroll
    for (int ni = 0; ni < 4; ++ni) {
      int colg = n0 + ni * 16 + lm;
      int h = colg >> 7, d = colg & (kD - 1);
      #pragma unroll
      for (int i = 0; i < 8; ++i) {
        int rowg = m0 + mi * 16 + i + (hf ? 8 : 0);
        int b = rowg >> 12, s = rowg & (kS - 1);
        Out[(((size_t)b * kH + h) * kS + s) * kD + d] = acc[mi * 4 + ni][i];
      }
    }
  }
}

// ---------------------------------------------------------------------------
// K2: LSH buckets. rotations is row-major [D=128, NH*R=512] == WMMA B operand.
// One wave handles a 16-row s-strip; argmax over concat[rot,-rot] per hash,
// first-occurrence tie-break, then cross-lane reduce over the 16-col group.
// ---------------------------------------------------------------------------
__global__ __launch_bounds__(128) void hash_kernel(
    const float* __restrict__ qk, const float* __restrict__ rot,
    int* __restrict__ buckets) {
  const int lane = threadIdx.x & 31;
  const int wave = threadIdx.x >> 5;
  const int lm = lane & 15, hf = lane >> 4;
  int tile = blockIdx.x * 4 + wave;      // B*H*(S/16) = 4096 tiles
  int stile = tile & ((kS / 16) - 1);
  int bh = tile / (kS / 16);
  int s0 = stile * 16;

  // Cache all A fragments for this 16-row strip (32 x f32x2 = 64 VGPRs).
  const f32x2* qp =
      (const f32x2*)(qk + ((size_t)bh * kS + s0 + lm) * kD + hf * 2);
  f32x2 afrag[32];
  #pragma unroll
  for (int j = 0; j < 32; ++j) afrag[j] = qp[2 * j];

  for (int n = 0; n < kNH; ++n) {
    float bpv[8], bnv[8]; int bpi[8], bni[8];
    #pragma unroll
    for (int i = 0; i < 8; ++i) { bpv[i] = -3.4e38f; bnv[i] = -3.4e38f; bpi[i] = 0; bni[i] = 0; }
    for (int ct = 0; ct < 8; ++ct) {
      const int col = n * 128 + ct * 16 + lm;
      f32x8 c = {};
      #pragma unroll
      for (int j = 0; j < 32; ++j) {
        int kb = j * 4 + hf * 2;
        f32x2 b;
        b.x = rot[(size_t)kb * 512 + col];
        b.y = rot[(size_t)(kb + 1) * 512 + col];
        c = wmma_f32(afrag[j], b, c);
      }
      int r = ct * 16 + lm;  // column index within this hash round
      #pragma unroll
      for (int i = 0; i < 8; ++i) {
        float v = c[i];
        if (v > bpv[i])  { bpv[i] = v;  bpi[i] = r; }
        if (-v > bnv[i]) { bnv[i] = -v; bni[i] = r; }
      }
    }
    #pragma unroll
    for (int m = 1; m <= 8; m <<= 1) {
      #pragma unroll
      for (int i = 0; i < 8; ++i) {
        float ov = __shfl_xor(bpv[i], m, 32);
        int   oi = __shfl_xor(bpi[i], m, 32);
        if (ov > bpv[i] || (ov == bpv[i] && oi < bpi[i])) { bpv[i] = ov; bpi[i] = oi; }
        float ow = __shfl_xor(bnv[i], m, 32);
        int   oj = __shfl_xor(bni[i], m, 32);
        if (ow > bnv[i] || (ow == bnv[i] && oj < bni[i])) { bnv[i] = ow; bni[i] = oj; }
      }
    }
    if (lm == 0) {
      #pragma unroll
      for (int i = 0; i < 8; ++i) {
        int row = i + (hf ? 8 : 0);
        int idx = (bnv[i] > bpv[i]) ? (bni[i] + 128) : bpi[i];  // concat index
        buckets[((size_t)bh * kNH + n) * kS + s0 + row] = idx + n * 256;
      }
    }
  }
}

// ---------------------------------------------------------------------------
// K3: stable counting sort per (b,h): 16384 keys in [0,1024).
// Keys staged in LDS; each thread owns 4 bins and scans keys in order.
// ---------------------------------------------------------------------------
__global__ __launch_bounds__(256) void sort_kernel(
    const int* __restrict__ buckets, int* __restrict__ sidx) {
  extern __shared__ int smem_sort[];
  int* keys = smem_sort;              // 16384
  int* hist = keys + kNSort;          // 1024
  int* part = hist + 1024;            // 256
  const int t = threadIdx.x;
  const int bh = blockIdx.x;
  const int* bk = buckets + (size_t)bh * kNSort;

  for (int i = t; i < 1024; i += 256) hist[i] = 0;
  __syncthreads();
  for (int i = t; i < kNSort; i += 256) {
    int k = bk[i];
    keys[i] = k;
    atomicAdd(&hist[k], 1);
  }
  __syncthreads();
  int h0 = hist[t * 4 + 0], h1 = hist[t * 4 + 1];
  int h2 = hist[t * 4 + 2], h3 = hist[t * 4 + 3];
  part[t] = h0 + h1 + h2 + h3;
  __syncthreads();
  if (t == 0) {
    int run = 0;
    for (int i = 0; i < 256; ++i) { int tmp = part[i]; part[i] = run; run += tmp; }
  }
  __syncthreads();
  int p0 = part[t];
  int p1 = p0 + h0, p2 = p1 + h1, p3 = p2 + h2;
  int* out = sidx + (size_t)bh * kNSort;
  const int b0 = t * 4;
  for (int i = 0; i < kNSort; ++i) {        // stable: i ascending
    int d = keys[i] - b0;
    if ((unsigned)d < 4u) {
      int pos;
      if (d == 0)      pos = p0++;
      else if (d == 1) pos = p1++;
      else if (d == 2) pos = p2++;
      else             pos = p3++;
      out[pos] = i;
    }
  }
}

// ---------------------------------------------------------------------------
// K4: chunked attention. One block (4 waves) per (b,h,chunk).
// gather q/v (prev+cur chunk) -> WMMA dots -> masks -> logsumexp ->
// WMMA P@V -> scatter-unsort via full sorted index.
// ---------------------------------------------------------------------------
__global__ __launch_bounds__(128) void attn_kernel(
    const float* __restrict__ qk, const float* __restrict__ v,
    const int* __restrict__ sidx, float* __restrict__ rout,
    float* __restrict__ rlog) {
  extern __shared__ float smem_attn[];
  float* gbuf  = smem_attn;            // 128 x GP  raw gathered q (prev|cur)
  float* vbuf  = gbuf + 128 * GP;      // 128 x GP  gathered v
  float* probs = vbuf + 128 * GP;      // 64 x GP
  float* scale = probs + 64 * GP;      // 128 key scales
  int*   kpos  = (int*)(scale + 128);  // 128 original positions (mod S)
  int*   ofull = kpos + 128;           // 64 full sorted indices for q rows

  const int t = threadIdx.x;
  const int lane = t & 31, wave = t >> 5;
  const int bh = blockIdx.x >> 8;
  const int c  = blockIdx.x & (kNChunks - 1);
  const int cprev = (c + kNChunks - 1) & (kNChunks - 1);
  const int* sbh = sidx + (size_t)bh * kNSort;

  {  // gather: row r = t (128 rows: 0-63 prev chunk, 64-127 current)
    int r = t;
    int j = (r < 64) ? (cprev * 64 + r) : (c * 64 + (r - 64));
    int full = sbh[j];
    int st = full & (kS - 1);
    kpos[r] = st;
    if (r >= 64) ofull[r - 64] = full;
    const float* qrow = qk + ((size_t)bh * kS + st) * kD;
    const float* vrow = v + ((size_t)bh * kS + st) * kD;
    float* gd = gbuf + r * GP;
    float* vd = vbuf + r * GP;
    float ss = 0.f;
    #pragma unroll 4
    for (int dd = 0; dd < kD; dd += 4) {
      float4 fq = *(const float4*)(qrow + dd);
      float4 fv = *(const float4*)(vrow + dd);
      *(float4*)(gd + dd) = fq;
      *(float4*)(vd + dd) = fv;
      ss += fq.x * fq.x + fq.y * fq.y + fq.z * fq.z + fq.w * fq.w;
    }
    // key scale: rsqrt(mean(q^2)+eps) * rsqrt(D)
    scale[r] = rsqrtf(ss * (1.0f / kD) + 1e-6f) * rsqrtf((float)kD);
  }
  __syncthreads();

  const int lm = lane & 15, hf = lane >> 4;
  const int m0 = wave * 16;  // this wave's q-row strip
  const float* arow = gbuf + (64 + m0 + lm) * GP;

  // dots = Q . G^T, scaled per key column, then masks
  for (int nt = 0; nt < 8; ++nt) {
    int n0 = nt * 16;
    const float* brow = gbuf + (n0 + lm) * GP;  // B[k][n] = gbuf[n][k]
    f32x8 cacc = {};
    #pragma unroll 4
    for (int j = 0; j < 32; ++j) {
      int kb = j * 4 + hf * 2;
      f32x2 a, b;
      a.x = arow[kb]; a.y = arow[kb + 1];
      b.x = brow[kb]; b.y = brow[kb + 1];
      cacc = wmma_f32(a, b, cacc);
    }
    float sc = scale[n0 + lm];
    int kp = kpos[n0 + lm];
    #pragma unroll
    for (int i = 0; i < 8; ++i) {
      int m = m0 + i + (hf ? 8 : 0);
      int qp = kpos[64 + m];
      float val = cacc[i] * sc;
      if (qp < kp)       val = -1e9f;   // causal mask
      else if (qp == kp) val = -1e5f;   // self mask
      probs[m * GP + n0 + lm] = val;
    }
  }
  __syncthreads();

  {  // logsumexp + exp, 2 lanes per row (64 cols each)
    int row = m0 + (lane >> 1);
    int c0 = (lane & 1) * 64;
    float* pr = probs + row * GP + c0;
    float mx = -3.4e38f;
    for (int kk = 0; kk < 64; ++kk) mx = fmaxf(mx, pr[kk]);
    mx = fmaxf(mx, __shfl_xor(mx, 1, 32));
    float sum = 0.f;
    for (int kk = 0; kk < 64; ++kk) sum += expf(pr[kk] - mx);
    sum += __shfl_xor(sum, 1, 32);
    float lg = mx + logf(sum);
    for (int kk = 0; kk < 64; ++kk) pr[kk] = expf(pr[kk] - lg);
    if ((lane & 1) == 0) rlog[(size_t)bh * kNSort + ofull[row]] = lg;
  }
  __syncthreads();

  // out = P @ V, scatter rows to unsorted position ofull[m]
  const float* prow = probs + (m0 + lm) * GP;
  for (int nt = 0; nt < 8; ++nt) {
    int n0 = nt * 16;
    f32x8 cacc = {};
    #pragma unroll 4
    for (int j = 0; j < 32; ++j) {
      int kb = j * 4 + hf * 2;
      f32x2 a, b;
      a.x = prow[kb]; a.y = prow[kb + 1];
      b.x = vbuf[kb * GP + n0 + lm];
      b.y = vbuf[(kb + 1) * GP + n0 + lm];
      cacc = wmma_f32(a, b, cacc);
    }
    #pragma unroll
    for (int i = 0; i < 8; ++i) {
      int m = m0 + i + (hf ? 8 : 0);
      rout[((size_t)bh * kNSort + ofull[m]) * kD + n0 + lm] = cacc[i];
    }
  }
}

// ---------------------------------------------------------------------------
// K5: combine hash rounds with softmax over per-round logsumexp; merge heads.
// ---------------------------------------------------------------------------
__global__ __launch_bounds__(256) void combine_kernel(
    const float* __restrict__ rout, const float* __restrict__ rlog,
    float* __restrict__ out) {
  size_t gid = (size_t)blockIdx.x * 256 + threadIdx.x;  // B*S*H*(D/4)
  int dg = (int)(gid & 31);
  size_t rem = gid >> 5;
  int h = (int)(rem & 7);
  rem >>= 3;
  int s = (int)(rem & (kS - 1));
  int b = (int)(rem >> 12);
  int bh = b * kH + h;
  const float* lg = rlog + (size_t)bh * kNSort;
  float l0 = lg[0 * kS + s], l1 = lg[1 * kS + s];
  float l2 = lg[2 * kS + s], l3 = lg[3 * kS + s];
  float mx = fmaxf(fmaxf(l0, l1), fmaxf(l2, l3));
  float w0 = expf(l0 - mx), w1 = expf(l1 - mx);
  float w2 = expf(l2 - mx), w3 = expf(l3 - mx);
  float inv = 1.f / (w0 + w1 + w2 + w3);
  w0 *= inv; w1 *= inv; w2 *= inv; w3 *= inv;
  const float* rb = rout + (size_t)bh * kNSort * kD;
  int d0 = dg * 4;
  float4 a0 = *(const float4*)(rb + (size_t)(0 * kS + s) * kD + d0);
  float4 a1 = *(const float4*)(rb + (size_t)(1 * kS + s) * kD + d0);
  float4 a2 = *(const float4*)(rb + (size_t)(2 * kS + s) * kD + d0);
  float4 a3 = *(const float4*)(rb + (size_t)(3 * kS + s) * kD + d0);
  float4 r;
  r.x = a0.x * w0 + a1.x * w1 + a2.x * w2 + a3.x * w3;
  r.y = a0.y * w0 + a1.y * w1 + a2.y * w2 + a3.y * w3;
  r.z = a0.z * w0 + a1.z * w1 + a2.z * w2 + a3.z * w3;
  r.w = a0.w * w0 + a1.w * w1 + a2.w * w2 + a3.w * w3;
  *(float4*)(out + ((size_t)b * kS + s) * kHid + h * kD + d0) = r;
}

// ---------------------------------------------------------------------------
extern "C" void kernel_launch(void* const* d_in, const int* in_sizes, int n_in,
                              void* d_out, int out_size, void* d_ws,
                              size_t ws_size, hipStream_t stream) {
  (void)in_sizes; (void)n_in; (void)out_size; (void)ws_size;
  const float* X   = (const float*)d_in[0];  // [B,S,HID]
  const float* Wqk = (const float*)d_in[1];  // [HID,HID]
  const float* Wv  = (const float*)d_in[2];  // [HID,HID]
  const float* Rot = (const float*)d_in[3];  // [D,NH,128] == [128,512] row-major
  float* out = (float*)d_out;

  // workspace layout (floats): ~200 MB total
  float* ws = (float*)d_ws;
  const size_t nBHSD = (size_t)kB * kH * kS * kD;   // 8,388,608
  float* qk_ws = ws;
  float* v_ws  = qk_ws + nBHSD;
  float* rout  = v_ws + nBHSD;                      // [B,H,NH*S,D]
  float* rlog  = rout + nBHSD * kNH;                // [B,H,NH*S]
  int* buckets = (int*)(rlog + (size_t)kB * kH * kNSort);
  int* sidx    = buckets + (size_t)kB * kH * kNSort;

  const int SORT_LDS = (kNSort + 1024 + 256) * 4;                       // ~70.7 KB
  const int ATTN_LDS = (128 * GP * 2 + 64 * GP + 128) * 4 + 192 * 4;    // ~170 KB

  (void)hipFuncSetAttribute((const void*)sort_kernel,
                            hipFuncAttributeMaxDynamicSharedMemorySize, SORT_LDS);
  (void)hipFuncSetAttribute((const void*)attn_kernel,
                            hipFuncAttributeMaxDynamicSharedMemorySize, ATTN_LDS);

  // K1: projections, 4096 wave-tiles (64x64 each) / 8 waves per block
  proj_kernel<<<512, 256, 0, stream>>>(X, Wqk, Wv, qk_ws, v_ws);
  // K2: LSH buckets, 4096 wave-strips / 4 waves per block
  hash_kernel<<<1024, 128, 0, stream>>>(qk_ws, Rot, buckets);
  // K3: stable sort per (b,h)
  sort_kernel<<<kB * kH, 256, SORT_LDS, stream>>>(buckets, sidx);
  // K4: chunked attention, one block per (b,h,chunk)
  attn_kernel<<<kB * kH * kNChunks, 128, ATTN_LDS, stream>>>(qk_ws, v_ws, sidx,
                                                             rout, rlog);
  // K5: round combine + head merge
  combine_kernel<<<(kB * kS * kH * (kD / 4)) / 256, 256, 0, stream>>>(rout, rlog,
                                                                      out);
}